// FederatedPPOAgent_50757923504226
// MI455X (gfx1250) — compile-verified
//
#include <hip/hip_runtime.h>
#include <hip/hip_bf16.h>

typedef __attribute__((ext_vector_type(2))) float v2f;
typedef __attribute__((ext_vector_type(8))) float v8f;

// ---------------------------------------------------------------------------
// Degree / norm precompute
// ---------------------------------------------------------------------------
__global__ __launch_bounds__(256) void k_fill1(float* p, int n) {
  int i = blockIdx.x * 256 + threadIdx.x;
  if (i < n) p[i] = 1.0f;   // self-loop contributes 1 to every node's degree
}

__global__ __launch_bounds__(256) void k_deg_acc(const long long* __restrict__ dst,
                                                 float* deg, long long E) {
  long long i = (long long)blockIdx.x * 256 + threadIdx.x;
  if (i < E) unsafeAtomicAdd(&deg[dst[i]], 1.0f);
}

__global__ __launch_bounds__(256) void k_rsqrt_inplace(float* p, int n) {
  int i = blockIdx.x * 256 + threadIdx.x;
  if (i < n) p[i] = rsqrtf(p[i]);
}

__global__ __launch_bounds__(256) void k_norm(const long long* __restrict__ src,
                                              const long long* __restrict__ dst,
                                              const float* __restrict__ dinv,
                                              float* __restrict__ norm, long long E) {
  long long i = (long long)blockIdx.x * 256 + threadIdx.x;
  if (i < E) norm[i] = dinv[src[i]] * dinv[dst[i]];
}

// ---------------------------------------------------------------------------
// WMMA GEMM: out[M,Nout] = act(A)[M,K] @ W[K,Nout] (+bias)(+relu)
// One wave computes a 32x32 macro-tile (2x2 grid of 16x16 WMMA tiles):
//  - 4 independent accumulator chains -> ILP across v_wmma latency
//  - A fragments loaded once per 2 output columns tiles, B once per 2 row tiles
// A 16x4 f32 frag: lanes 0-15 hold K=+0,+1; lanes 16-31 hold K=+2,+3 (b64 load).
// B/C/D: row-striped across lanes; VGPR v -> rows v (lanes 0-15) / v+8 (16-31).
// M, Nout multiples of 32; K multiple of 4. Tile guard is wave-uniform so
// EXEC is all-ones at every v_wmma (ISA requirement).
// ---------------------------------------------------------------------------
template <int RELU_IN, int RELU_OUT, int HAS_BIAS>
__global__ __launch_bounds__(256) void k_gemm_wmma(const float* __restrict__ A,
                                                   const float* __restrict__ W,
                                                   const float* __restrict__ bias,
                                                   float* __restrict__ out,
                                                   int M, int K, int Nout) {
  const int wave = threadIdx.x >> 5;
  const int lane = threadIdx.x & 31;
  const int half = lane >> 4;      // 0: lanes 0-15, 1: lanes 16-31
  const int l    = lane & 15;

  const int tilesN = Nout >> 5;                 // 32-wide macro-tiles
  const int totalTiles = (M >> 5) * tilesN;
  const int tile = blockIdx.x * 8 + wave;
  if (tile >= totalTiles) return;               // wave-uniform -> EXEC all ones

  const int m0 = (tile / tilesN) << 5;
  const int n0 = (tile % tilesN) << 5;

  const float* __restrict__ ar0 = A + (long long)(m0 + l) * K + 2 * half;
  const float* __restrict__ ar1 = ar0 + (long long)16 * K;

  v8f acc00 = {}, acc01 = {}, acc10 = {}, acc11 = {};

#pragma unroll 2
  for (int k0 = 0; k0 < K; k0 += 4) {
    v2f a0 = *(const v2f*)(ar0 + k0);
    v2f a1 = *(const v2f*)(ar1 + k0);
    if (RELU_IN) {
      a0.x = fmaxf(a0.x, 0.0f); a0.y = fmaxf(a0.y, 0.0f);
      a1.x = fmaxf(a1.x, 0.0f); a1.y = fmaxf(a1.y, 0.0f);
    }
    const float* wr = W + (long long)(k0 + 2 * half) * Nout + n0 + l;
    v2f b0, b1;
    b0.x = wr[0];       b0.y = wr[Nout];        // K=+0,+1 (or +2,+3), col n0+l
    b1.x = wr[16];      b1.y = wr[Nout + 16];   // col n0+16+l
    acc00 = __builtin_amdgcn_wmma_f32_16x16x4_f32(false, a0, false, b0, (short)0, acc00, false, false);
    acc01 = __builtin_amdgcn_wmma_f32_16x16x4_f32(false, a0, false, b1, (short)0, acc01, false, false);
    acc10 = __builtin_amdgcn_wmma_f32_16x16x4_f32(false, a1, false, b0, (short)0, acc10, false, false);
    acc11 = __builtin_amdgcn_wmma_f32_16x16x4_f32(false, a1, false, b1, (short)0, acc11, false, false);
  }

  const int col0 = n0 + l;
  const int col1 = col0 + 16;
  const float bv0 = HAS_BIAS ? bias[col0] : 0.0f;
  const float bv1 = HAS_BIAS ? bias[col1] : 0.0f;
#pragma unroll
  for (int v = 0; v < 8; ++v) {
    const int r0 = m0 + v + 8 * half;
    const int r1 = r0 + 16;
    float v00 = acc00[v] + bv0, v01 = acc01[v] + bv1;
    float v10 = acc10[v] + bv0, v11 = acc11[v] + bv1;
    if (RELU_OUT) {
      v00 = fmaxf(v00, 0.0f); v01 = fmaxf(v01, 0.0f);
      v10 = fmaxf(v10, 0.0f); v11 = fmaxf(v11, 0.0f);
    }
    out[(long long)r0 * Nout + col0] = v00;
    out[(long long)r0 * Nout + col1] = v01;
    out[(long long)r1 * Nout + col0] = v10;
    out[(long long)r1 * Nout + col1] = v11;
  }
}

// ---------------------------------------------------------------------------
// agg[i,c] = XW[i,c] * dinv[i]^2 + bias[c]   (self-loop term + bias init)
// ---------------------------------------------------------------------------
__global__ __launch_bounds__(256) void k_self_init(const float* __restrict__ XW,
                                                   const float* __restrict__ dinv,
                                                   const float* __restrict__ bias,
                                                   float* __restrict__ agg,
                                                   int cshift, int total) {
  int i = blockIdx.x * 256 + threadIdx.x;
  if (i >= total) return;
  const int node = i >> cshift;
  const int c    = i & ((1 << cshift) - 1);
  const float dv = dinv[node];
  agg[i] = XW[i] * dv * dv + bias[c];
}

// ---------------------------------------------------------------------------
// Edge scatter: agg[dst] += norm[e] * XW[src].
// One thread = one edge x 4 features; a full wave covers one edge's 128 (or
// two edges' 64) features, so src/dst/norm loads are wave-uniform broadcasts.
// Atomics resolve at L2 (feature matrix is 51MB < 192MB L2).
// ---------------------------------------------------------------------------
__global__ __launch_bounds__(256) void k_scatter(const long long* __restrict__ src,
                                                 const long long* __restrict__ dst,
                                                 const float* __restrict__ norm,
                                                 const float* __restrict__ XW,
                                                 float* __restrict__ agg,
                                                 int C, int cshift, long long E) {
  const long long idx = (long long)blockIdx.x * 256 + threadIdx.x;
  const long long total = E << cshift;
  if (idx >= total) return;
  const long long e  = idx >> cshift;
  const int       ch = (int)(idx & ((1 << cshift) - 1));
  const long long s = src[e];
  const long long d = dst[e];
  const float w = norm[e];
  const float4 m = *((const float4*)(XW + s * C) + ch);
  float* ap = agg + d * C + ch * 4;
  unsafeAtomicAdd(ap + 0, m.x * w);
  unsafeAtomicAdd(ap + 1, m.y * w);
  unsafeAtomicAdd(ap + 2, m.z * w);
  unsafeAtomicAdd(ap + 3, m.w * w);
}

// ---------------------------------------------------------------------------
// Heads: 64 -> 8 logits + softmax (cols 0..7), 64 -> 1 value (col 8)
// ---------------------------------------------------------------------------
__global__ __launch_bounds__(256) void k_actor_head(const float* __restrict__ A2,
                                                    const float* __restrict__ Wa3,
                                                    const float* __restrict__ ba3,
                                                    float* __restrict__ out, int Nn) {
  int i = blockIdx.x * 256 + threadIdx.x;
  if (i >= Nn) return;
  const float* h = A2 + (long long)i * 64;
  float logit[8];
#pragma unroll
  for (int j = 0; j < 8; ++j) logit[j] = ba3[j];
  for (int k = 0; k < 64; ++k) {
    const float hv = h[k];
#pragma unroll
    for (int j = 0; j < 8; ++j) logit[j] += hv * Wa3[k * 8 + j];
  }
  float mx = logit[0];
#pragma unroll
  for (int j = 1; j < 8; ++j) mx = fmaxf(mx, logit[j]);
  float sum = 0.0f;
#pragma unroll
  for (int j = 0; j < 8; ++j) { logit[j] = __expf(logit[j] - mx); sum += logit[j]; }
  const float inv = 1.0f / sum;
#pragma unroll
  for (int j = 0; j < 8; ++j) out[(long long)i * 9 + j] = logit[j] * inv;
}

__global__ __launch_bounds__(256) void k_critic_head(const float* __restrict__ C2,
                                                     const float* __restrict__ Wc3,
                                                     const float* __restrict__ bc3,
                                                     float* __restrict__ out, int Nn) {
  int i = blockIdx.x * 256 + threadIdx.x;
  if (i >= Nn) return;
  const float* h = C2 + (long long)i * 64;
  float v = bc3[0];
  for (int k = 0; k < 64; ++k) v += h[k] * Wc3[k];
  out[(long long)i * 9 + 8] = v;
}

// ---------------------------------------------------------------------------
// Launcher
// ---------------------------------------------------------------------------
static inline int cdiv_ll(long long a, long long b) { return (int)((a + b - 1) / b); }

extern "C" void kernel_launch(void* const* d_in, const int* in_sizes, int n_in,
                              void* d_out, int out_size, void* d_ws, size_t ws_size,
                              hipStream_t stream) {
  (void)n_in; (void)out_size; (void)ws_size;

  const float*      x    = (const float*)d_in[0];
  const long long*  eidx = (const long long*)d_in[1];
  const float *W1 = (const float*)d_in[2],  *b1  = (const float*)d_in[3];
  const float *W2 = (const float*)d_in[4],  *b2  = (const float*)d_in[5];
  const float *W3 = (const float*)d_in[6],  *b3  = (const float*)d_in[7];
  const float *Wa1= (const float*)d_in[8],  *ba1 = (const float*)d_in[9];
  const float *Wa2= (const float*)d_in[10], *ba2 = (const float*)d_in[11];
  const float *Wa3= (const float*)d_in[12], *ba3 = (const float*)d_in[13];
  const float *Wc1= (const float*)d_in[14], *bc1 = (const float*)d_in[15];
  const float *Wc2= (const float*)d_in[16], *bc2 = (const float*)d_in[17];
  const float *Wc3= (const float*)d_in[18], *bc3 = (const float*)d_in[19];

  const int       Nn = in_sizes[0] / 128;       // 100000 (multiple of 32)
  const long long E  = (long long)in_sizes[1] / 2;
  const long long* src = eidx;
  const long long* dst = eidx + E;

  float* ws   = (float*)d_ws;
  size_t off  = 0;
  float* dinv = ws + off; off += ((size_t)Nn + 255) & ~(size_t)255;
  float* norm = ws + off; off += ((size_t)E  + 255) & ~(size_t)255;
  float* bufX = ws + off; off += (size_t)Nn * 128;   // XW / A1 / C1
  float* bufH = ws + off;                            // AGG / H (+A2/C2 upper half)
  float* bufT = bufH + (size_t)Nn * 64;              // A2 / C2 (N x 64)
  float* outp = (float*)d_out;

  // ---- degrees & per-edge norm --------------------------------------------
  k_fill1        <<<cdiv_ll(Nn, 256), 256, 0, stream>>>(dinv, Nn);
  k_deg_acc      <<<cdiv_ll(E, 256), 256, 0, stream>>>(dst, dinv, E);
  k_rsqrt_inplace<<<cdiv_ll(Nn, 256), 256, 0, stream>>>(dinv, Nn);
  k_norm         <<<cdiv_ll(E, 256), 256, 0, stream>>>(src, dst, dinv, norm, E);

  auto blocks = [&](int Nout) { return cdiv_ll((long long)(Nn / 32) * (Nout / 32), 8); };

  // ---- GCN layer 1: AGG1 = self+b1+scatter(X@W1); relu deferred to next load
  k_gemm_wmma<0,0,0><<<blocks(128), 256, 0, stream>>>(x, W1, nullptr, bufX, Nn, 128, 128);
  k_self_init<<<cdiv_ll((long long)Nn * 128, 256), 256, 0, stream>>>(bufX, dinv, b1, bufH, 7, Nn * 128);
  k_scatter  <<<cdiv_ll(E << 5, 256), 256, 0, stream>>>(src, dst, norm, bufX, bufH, 128, 5, E);

  // ---- GCN layer 2 ---------------------------------------------------------
  k_gemm_wmma<1,0,0><<<blocks(128), 256, 0, stream>>>(bufH, W2, nullptr, bufX, Nn, 128, 128);
  k_self_init<<<cdiv_ll((long long)Nn * 128, 256), 256, 0, stream>>>(bufX, dinv, b2, bufH, 7, Nn * 128);
  k_scatter  <<<cdiv_ll(E << 5, 256), 256, 0, stream>>>(src, dst, norm, bufX, bufH, 128, 5, E);

  // ---- GCN layer 3 (no relu on output) ------------------------------------
  k_gemm_wmma<1,0,0><<<blocks(64), 256, 0, stream>>>(bufH, W3, nullptr, bufX, Nn, 128, 64);
  k_self_init<<<cdiv_ll((long long)Nn * 64, 256), 256, 0, stream>>>(bufX, dinv, b3, bufH, 6, Nn * 64);
  k_scatter  <<<cdiv_ll(E << 4, 256), 256, 0, stream>>>(src, dst, norm, bufX, bufH, 64, 4, E);
  // H = bufH[0 .. Nn*64)

  // ---- actor head ----------------------------------------------------------
  k_gemm_wmma<0,1,1><<<blocks(128), 256, 0, stream>>>(bufH, Wa1, ba1, bufX, Nn, 64, 128);  // A1
  k_gemm_wmma<0,1,1><<<blocks(64),  256, 0, stream>>>(bufX, Wa2, ba2, bufT, Nn, 128, 64);  // A2
  k_actor_head<<<cdiv_ll(Nn, 256), 256, 0, stream>>>(bufT, Wa3, ba3, outp, Nn);

  // ---- critic head ---------------------------------------------------------
  k_gemm_wmma<0,1,1><<<blocks(128), 256, 0, stream>>>(bufH, Wc1, bc1, bufX, Nn, 64, 128);  // C1
  k_gemm_wmma<0,1,1><<<blocks(64),  256, 0, stream>>>(bufX, Wc2, bc2, bufT, Nn, 128, 64);  // C2
  k_critic_head<<<cdiv_ll(Nn, 256), 256, 0, stream>>>(bufT, Wc3, bc3, outp, Nn);
}